// AMSGP_69380901699622
// MI455X (gfx1250) — compile-verified
//
#include <hip/hip_runtime.h>
#include <math.h>

#define N_NODES 8192
#define DEMB    256
#define XINC    128
#define DCAT    384
#define NHEAD   4
#define ODIM    128
#define HO      512
#define NEDGE   262144
#define CHUNKSZ 65536

#define KVST 396   // LDS row stride (dwords) for K/V tiles: conflict-free, 16B aligned
#define GPST 36    // LDS row stride (dwords) for GEMM panels

typedef __attribute__((ext_vector_type(2))) float v2f;
typedef __attribute__((ext_vector_type(8))) float v8f;

static __device__ __forceinline__ v8f v8f_zero() {
  v8f z = {0.f,0.f,0.f,0.f,0.f,0.f,0.f,0.f};
  return z;
}
static __device__ __forceinline__ v2f make_v2f(float a, float b) {
  v2f r; r.x = a; r.y = b; return r;
}
// D = A(16x4 f32) * B(4x16 f32) + C(16x16 f32), full f32 precision matrix core
static __device__ __forceinline__ v8f wmma_f32x4(v2f a, v2f b, v8f c) {
  return __builtin_amdgcn_wmma_f32_16x16x4_f32(false, a, false, b, (short)0, c, false, false);
}

// CDNA5 async DMA: global -> LDS without VGPR transit; tracked by ASYNCcnt.
static __device__ __forceinline__ void async_b128(void* lds, const void* gaddr) {
  uint32_t l = (uint32_t)(size_t)lds;          // LDS offset (addrspace(3))
  uint64_t g = (uint64_t)(size_t)gaddr;
  asm volatile("global_load_async_to_lds_b128 %0, %1, off"
               :: "v"(l), "v"(g) : "memory");
}
static __device__ __forceinline__ void wait_async0() {
  asm volatile("s_wait_asynccnt 0x0" ::: "memory");
}

// ---------------------------------------------------------------------------
// cat = [tile(x,128) | E_all[ids]]  -> [N, 384]
// ---------------------------------------------------------------------------
__global__ void build_cat_kernel(const int* __restrict__ ids, const float* __restrict__ x,
                                 const float* __restrict__ E_all, float* __restrict__ cat) {
  int t = blockIdx.x * 256 + threadIdx.x;
  if (t >= N_NODES * DCAT) return;
  int n = t / DCAT, c = t - n * DCAT;
  cat[t] = (c < XINC) ? x[n] : E_all[(size_t)ids[n] * DEMB + (c - XINC)];
}

// ---------------------------------------------------------------------------
// C[M,N] = A[M,K] @ W[N,K]^T + bias  (x @ W.T).  Block = 64x64 tile, 4 waves.
// K panels of 32 in double-buffered LDS, filled by async-to-LDS DMA issued at
// the top of each iteration (next panel) and waited after the 32 wmmas.
// ---------------------------------------------------------------------------
__global__ void __launch_bounds__(128, 1)
gemm_xwt_kernel(const float* __restrict__ A, const float* __restrict__ W,
                const float* __restrict__ bias, float* __restrict__ C,
                int K, int ldc) {
  __shared__ float Ap[2][64][GPST];
  __shared__ float Wp[2][64][GPST];
  const int tid  = threadIdx.x;
  const int lane = tid & 31;
  const int w    = tid >> 5;
  const int half = lane >> 4;
  const int l16  = lane & 15;
  const int n0 = blockIdx.x * 64;
  const int m0 = blockIdx.y * 64;

  v8f acc[4];
#pragma unroll
  for (int t = 0; t < 4; ++t) acc[t] = v8f_zero();

  // panel = 64 rows x 32 cols = 512 float4; 4 async b128 per matrix per thread
#pragma unroll
  for (int i = 0; i < 4; ++i) {
    int idx = tid + 128 * i;
    int row = idx >> 3, cq = (idx & 7) << 2;
    async_b128(&Ap[0][row][cq], A + (size_t)(m0 + row) * K + cq);
    async_b128(&Wp[0][row][cq], W + (size_t)(n0 + row) * K + cq);
  }
  wait_async0();
  __syncthreads();

  int buf = 0;
  for (int kb = 0; kb < K; kb += 32) {
    if (kb + 32 < K) {  // async prefetch of next panel into the other buffer
#pragma unroll
      for (int i = 0; i < 4; ++i) {
        int idx = tid + 128 * i;
        int row = idx >> 3, cq = (idx & 7) << 2;
        async_b128(&Ap[buf ^ 1][row][cq], A + (size_t)(m0 + row) * K + kb + 32 + cq);
        async_b128(&Wp[buf ^ 1][row][cq], W + (size_t)(n0 + row) * K + kb + 32 + cq);
      }
    }
    // compute current panel entirely from LDS
    const float* Apb = &Ap[buf][0][0];
    const float* Wpb = &Wp[buf][0][0];
#pragma unroll
    for (int k2 = 0; k2 < 32; k2 += 4) {
      v2f a = *(const v2f*)(Apb + (w * 16 + l16) * GPST + k2 + 2 * half);
#pragma unroll
      for (int t = 0; t < 4; ++t) {
        v2f b = *(const v2f*)(Wpb + (t * 16 + l16) * GPST + k2 + 2 * half);
        acc[t] = wmma_f32x4(a, b, acc[t]);
      }
    }
    wait_async0();      // own prefetch complete (no-op on last panel)
    __syncthreads();    // all waves done reading buf + all prefetch visible
    buf ^= 1;
  }

#pragma unroll
  for (int t = 0; t < 4; ++t) {
    const float bv = bias[n0 + t * 16 + l16];
#pragma unroll
    for (int r = 0; r < 8; ++r)
      C[(size_t)(m0 + w * 16 + half * 8 + r) * ldc + n0 + t * 16 + l16] = acc[t][r] + bv;
  }
}

// ---------------------------------------------------------------------------
// Fused flash attention: out[N,384] = softmax(q@k.T * 384^-0.5) @ v
// qkv layout: [N, 1152] = [q | k | v].  One wave = 16 query rows; the 4 waves
// of a block share double-buffered LDS K/V tiles filled by async-to-LDS DMA
// (ASYNCcnt is independent of the Q loads' LOADcnt, so no ordering coupling).
// ---------------------------------------------------------------------------
__global__ void __launch_bounds__(128, 1)
flash_attn_kernel(const float* __restrict__ qkv, float* __restrict__ out) {
  __shared__ float Kt[2][16][KVST];
  __shared__ float Vt[2][16][KVST];
  __shared__ float Pt[4][16][16];
  const int tid  = threadIdx.x;
  const int w    = tid >> 5;
  const int lane = tid & 31;
  const int half = lane >> 4;
  const int l16  = lane & 15;
  const int m0   = (blockIdx.x * 4 + w) * 16;
  const int ld   = 3 * DCAT;
  const float scale = 0.05103103630798288f;  // 384^-0.5

  const float* __restrict__ qrow  = qkv + (size_t)(m0 + l16) * ld + 2 * half;
  const float* __restrict__ kbase = qkv + DCAT;
  const float* __restrict__ vbase = qkv + 2 * DCAT;

  v8f acc[24];
#pragma unroll
  for (int t = 0; t < 24; ++t) acc[t] = v8f_zero();
  float mstat[8], lstat[8];
#pragma unroll
  for (int r = 0; r < 8; ++r) { mstat[r] = -INFINITY; lstat[r] = 0.0f; }

  // stage tile 0: 16 keys x 384 = 1536 float4 per matrix, 12 asyncs per thread
#pragma unroll
  for (int i = 0; i < 12; ++i) {
    int idx = tid + 128 * i;
    int row = idx / 96, cq = (idx - row * 96) * 4;
    async_b128(&Kt[0][row][cq], kbase + (size_t)row * ld + cq);
    async_b128(&Vt[0][row][cq], vbase + (size_t)row * ld + cq);
  }
  wait_async0();
  __syncthreads();

  int buf = 0;
  for (int key0 = 0; key0 < N_NODES; key0 += 16) {
    const int nk = key0 + 16;
    if (nk < N_NODES) {  // async prefetch of next K/V tile into other buffer
#pragma unroll
      for (int i = 0; i < 12; ++i) {
        int idx = tid + 128 * i;
        int row = idx / 96, cq = (idx - row * 96) * 4;
        async_b128(&Kt[buf ^ 1][row][cq], kbase + (size_t)(nk + row) * ld + cq);
        async_b128(&Vt[buf ^ 1][row][cq], vbase + (size_t)(nk + row) * ld + cq);
      }
    }
    const float* Ktb = &Kt[buf][0][0];
    const float* Vtb = &Vt[buf][0][0];

    // ---- S = Q_tile @ K_tile^T (K frags from LDS, Q chunk-double-buffered) ----
    v8f S0 = v8f_zero(), S1 = v8f_zero();
    v2f qa[2][8];
#pragma unroll
    for (int j = 0; j < 8; ++j) qa[0][j] = *(const v2f*)(qrow + j * 4);
#pragma unroll
    for (int c = 0; c < 12; ++c) {
      const int cur = c & 1;
      if (c < 11) {
#pragma unroll
        for (int j = 0; j < 8; ++j)
          qa[cur ^ 1][j] = *(const v2f*)(qrow + (c + 1) * 32 + j * 4);
      }
      const int kb = c * 32;
#pragma unroll
      for (int j = 0; j < 8; j += 2) {
        v2f b0 = *(const v2f*)(Ktb + l16 * KVST + kb + j * 4 + 2 * half);
        S0 = wmma_f32x4(qa[cur][j], b0, S0);
        v2f b1 = *(const v2f*)(Ktb + l16 * KVST + kb + (j + 1) * 4 + 2 * half);
        S1 = wmma_f32x4(qa[cur][j + 1], b1, S1);
      }
    }
    v8f S = S0 + S1;

    // ---- online softmax; row r of this half lives in VGPR r across 16 lanes ----
    float rs[8];
#pragma unroll
    for (int r = 0; r < 8; ++r) {
      float s = S[r] * scale;
      float mx = s;
#pragma unroll
      for (int mk = 1; mk < 16; mk <<= 1) mx = fmaxf(mx, __shfl_xor(mx, mk, 32));
      float mnew = fmaxf(mstat[r], mx);
      float p  = __expf(s - mnew);
      float ps = p;
#pragma unroll
      for (int mk = 1; mk < 16; mk <<= 1) ps += __shfl_xor(ps, mk, 32);
      float corr = __expf(mstat[r] - mnew);
      lstat[r] = lstat[r] * corr + ps;
      mstat[r] = mnew;
      rs[r] = corr;
      Pt[w][half * 8 + r][l16] = p;                 // C-layout -> LDS
    }
#pragma unroll
    for (int t = 0; t < 24; ++t)
#pragma unroll
      for (int r = 0; r < 8; ++r) acc[t][r] *= rs[r];

    // P back in A-layout (same-wave LDS is in-order)
    v2f pa[4];
#pragma unroll
    for (int j = 0; j < 4; ++j) {
      pa[j].x = Pt[w][l16][4 * j + 2 * half];
      pa[j].y = Pt[w][l16][4 * j + 2 * half + 1];
    }

    // ---- O += P @ V_tile (V frags from LDS) ----
#pragma unroll
    for (int t = 0; t < 24; ++t) {
#pragma unroll
      for (int j = 0; j < 4; ++j) {
        const int kr = 4 * j + 2 * half;
        v2f b = make_v2f(Vtb[kr * KVST + t * 16 + l16],
                         Vtb[(kr + 1) * KVST + t * 16 + l16]);
        acc[t] = wmma_f32x4(pa[j], b, acc[t]);
      }
    }

    wait_async0();      // own tile prefetch complete (no-op on last tile)
    __syncthreads();    // all waves: prefetch visible + done reading buf
    buf ^= 1;
  }

#pragma unroll
  for (int r = 0; r < 8; ++r) {
    float inv = 1.0f / lstat[r];
    float* __restrict__ orow = out + (size_t)(m0 + half * 8 + r) * DCAT + l16;
#pragma unroll
    for (int t = 0; t < 24; ++t) orow[t * 16] = acc[t][r] * inv;
  }
}

// ---------------------------------------------------------------------------
// LayerNorm(256) + SiLU + residual add of E_i (= cat[:,128:])
// ---------------------------------------------------------------------------
__global__ void ln_silu_add_kernel(const float* __restrict__ xin, const float* __restrict__ cat,
                                   const float* __restrict__ g, const float* __restrict__ b,
                                   float* __restrict__ out) {
  __shared__ float red[DEMB];
  const int n = blockIdx.x, c = threadIdx.x;
  const size_t idx = (size_t)n * DEMB + c;
  float v = xin[idx];
  red[c] = v; __syncthreads();
  for (int s = DEMB / 2; s > 0; s >>= 1) { if (c < s) red[c] += red[c + s]; __syncthreads(); }
  const float mu = red[0] * (1.0f / DEMB);
  __syncthreads();
  const float d = v - mu;
  red[c] = d * d; __syncthreads();
  for (int s = DEMB / 2; s > 0; s >>= 1) { if (c < s) red[c] += red[c + s]; __syncthreads(); }
  const float var = red[0] * (1.0f / DEMB);
  float y = d * rsqrtf(var + 1e-5f) * g[c] + b[c];
  y = y / (1.0f + __expf(-y));                     // SiLU
  out[idx] = y + cat[(size_t)n * DCAT + XINC + c]; // + E_i
}

// ---------------------------------------------------------------------------
// alpha[n,h] = dot(h[n, h*128 : h*128+128], attn[h, :])
// ---------------------------------------------------------------------------
__global__ void alpha_kernel(const float* __restrict__ h, const float* __restrict__ attn,
                             float* __restrict__ alpha) {
  int t = blockIdx.x * 256 + threadIdx.x;
  if (t >= N_NODES * NHEAD) return;
  int hd = t & 3, n = t >> 2;
  const float* hr = h + (size_t)n * HO + hd * ODIM;
  const float* ar = attn + hd * ODIM;
  float s = 0.0f;
  for (int o = 0; o < ODIM; ++o) s += hr[o] * ar[o];
  alpha[t] = s;
}

// ---------------------------------------------------------------------------
// a[e,h] = (alpha_i[dst,h] + alpha_j[src,h]) * edge_attr[e]
// ---------------------------------------------------------------------------
__global__ void edge_logits_kernel(const int* __restrict__ ei, const float* __restrict__ ea,
                                   const float* __restrict__ ai, const float* __restrict__ aj,
                                   float* __restrict__ a) {
  int e = blockIdx.x * 256 + threadIdx.x;
  if (e >= NEDGE) return;
  int s = ei[e], d = ei[NEDGE + e];
  float wv = ea[e];
#pragma unroll
  for (int h = 0; h < NHEAD; ++h)
    a[(size_t)e * NHEAD + h] = (ai[(size_t)d * NHEAD + h] + aj[(size_t)s * NHEAD + h]) * wv;
}

// ---------------------------------------------------------------------------
// softmax over 65536-edge chunks, per (chunk, head); in-place on a[E,4]
// ---------------------------------------------------------------------------
__global__ void chunk_softmax_kernel(float* __restrict__ a) {
  __shared__ float red[1024];
  const int ck = blockIdx.x >> 2;
  const int hd = blockIdx.x & 3;
  float* __restrict__ base = a + (size_t)ck * CHUNKSZ * NHEAD + hd;
  const int t = threadIdx.x;
  float mx = -INFINITY;
  for (int i = t; i < CHUNKSZ; i += 1024) mx = fmaxf(mx, base[(size_t)i * NHEAD]);
  red[t] = mx; __syncthreads();
  for (int s = 512; s > 0; s >>= 1) { if (t < s) red[t] = fmaxf(red[t], red[t + s]); __syncthreads(); }
  mx = red[0]; __syncthreads();
  float sm = 0.0f;
  for (int i = t; i < CHUNKSZ; i += 1024) {
    float p = __expf(base[(size_t)i * NHEAD] - mx);
    base[(size_t)i * NHEAD] = p;
    sm += p;
  }
  red[t] = sm; __syncthreads();
  for (int s = 512; s > 0; s >>= 1) { if (t < s) red[t] += red[t + s]; __syncthreads(); }
  const float inv = 1.0f / red[0];
  for (int i = t; i < CHUNKSZ; i += 1024) base[(size_t)i * NHEAD] *= inv;
}

// ---------------------------------------------------------------------------
// m[e,:] = sum_h a[e,h] * ht[src[e], h, :]; atomically scatter-add to agg[dst]
// ---------------------------------------------------------------------------
__global__ void edge_aggregate_kernel(const int* __restrict__ ei, const float* __restrict__ a,
                                      const float* __restrict__ ht, float* __restrict__ agg) {
  const int e = blockIdx.x;
  const int c = threadIdx.x;
  const int s = ei[e], d = ei[NEDGE + e];
  const float a0 = a[(size_t)e * 4 + 0], a1 = a[(size_t)e * 4 + 1];
  const float a2 = a[(size_t)e * 4 + 2], a3 = a[(size_t)e * 4 + 3];
  const float* hr = ht + (size_t)s * HO;
  float m = a0 * hr[c] + a1 * hr[ODIM + c] + a2 * hr[2 * ODIM + c] + a3 * hr[3 * ODIM + c];
  unsafeAtomicAdd(&agg[(size_t)d * ODIM + c], m);
}

// ---------------------------------------------------------------------------
// out = LayerNorm( agg1 + 0.5*(s0+s1) ) over 128 features
// ---------------------------------------------------------------------------
__global__ void final_ln_kernel(const float* __restrict__ agg1, const float* __restrict__ s0,
                                const float* __restrict__ s1, const float* __restrict__ g,
                                const float* __restrict__ b, float* __restrict__ out) {
  __shared__ float red[ODIM];
  const int n = blockIdx.x, c = threadIdx.x;
  const size_t idx = (size_t)n * ODIM + c;
  float v = agg1[idx] + 0.5f * (s0[idx] + s1[idx]);
  red[c] = v; __syncthreads();
  for (int s = ODIM / 2; s > 0; s >>= 1) { if (c < s) red[c] += red[c + s]; __syncthreads(); }
  const float mu = red[0] * (1.0f / ODIM);
  __syncthreads();
  const float d = v - mu;
  red[c] = d * d; __syncthreads();
  for (int s = ODIM / 2; s > 0; s >>= 1) { if (c < s) red[c] += red[c + s]; __syncthreads(); }
  const float var = red[0] * (1.0f / ODIM);
  out[idx] = d * rsqrtf(var + 1e-5f) * g[c] + b[c];
}

// ---------------------------------------------------------------------------
static inline void gemm_xwt(const float* A, const float* W, const float* bias, float* C,
                            int M, int N, int K, hipStream_t stream) {
  dim3 grid(N / 64, M / 64);
  gemm_xwt_kernel<<<grid, 128, 0, stream>>>(A, W, bias, C, K, N);
}

extern "C" void kernel_launch(void* const* d_in, const int* in_sizes, int n_in,
                              void* d_out, int out_size, void* d_ws, size_t ws_size,
                              hipStream_t stream) {
  const int*   ids    = (const int*)  d_in[0];
  const float* x      = (const float*)d_in[1];
  const float* eattr  = (const float*)d_in[2];
  const int*   eidx   = (const int*)  d_in[3];
  const float* E_all  = (const float*)d_in[4];
  const float* qkv_W  = (const float*)d_in[5];
  const float* qkv_b  = (const float*)d_in[6];
  const float* proj_W = (const float*)d_in[7];
  const float* proj_b = (const float*)d_in[8];
  const float* lin_W  = (const float*)d_in[9];
  const float* lin_b  = (const float*)d_in[10];
  const float* ln1_g  = (const float*)d_in[11];
  const float* ln1_b  = (const float*)d_in[12];
  const float* Wi0 = (const float*)d_in[13]; const float* bi0 = (const float*)d_in[14];
  const float* Wj0 = (const float*)d_in[15]; const float* bj0 = (const float*)d_in[16];
  const float* Wt0 = (const float*)d_in[17]; const float* bt0 = (const float*)d_in[18];
  const float* attn0 = (const float*)d_in[19];
  const float* Wi1 = (const float*)d_in[20]; const float* bi1 = (const float*)d_in[21];
  const float* Wj1 = (const float*)d_in[22]; const float* bj1 = (const float*)d_in[23];
  const float* Wt1 = (const float*)d_in[24]; const float* bt1 = (const float*)d_in[25];
  const float* attn1 = (const float*)d_in[26];
  const float* sk0_W = (const float*)d_in[27]; const float* sk0_b = (const float*)d_in[28];
  const float* sk1_W = (const float*)d_in[29]; const float* sk1_b = (const float*)d_in[30];
  const float* nrm_g = (const float*)d_in[31]; const float* nrm_b = (const float*)d_in[32];
  float* out = (float*)d_out;

  float* ws = (float*)d_ws;
  float* cat    = ws; ws += (size_t)N_NODES * DCAT;
  float* qkv    = ws; ws += (size_t)N_NODES * 3 * DCAT;
  float* att_o  = ws; ws += (size_t)N_NODES * DCAT;
  float* proj_o = ws; ws += (size_t)N_NODES * DCAT;
  float* lin_o  = ws; ws += (size_t)N_NODES * DEMB;
  float* emb0   = ws; ws += (size_t)N_NODES * DEMB;
  float* hi     = ws; ws += (size_t)N_NODES * HO;
  float* hj     = ws; ws += (size_t)N_NODES * HO;
  float* ht     = ws; ws += (size_t)N_NODES * HO;
  float* ai     = ws; ws += (size_t)N_NODES * NHEAD;
  float* aj     = ws; ws += (size_t)N_NODES * NHEAD;
  float* aedge  = ws; ws += (size_t)NEDGE * NHEAD;
  float* agg0   = ws; ws += (size_t)N_NODES * ODIM;
  float* agg1   = ws; ws += (size_t)N_NODES * ODIM;
  float* s0     = ws; ws += (size_t)N_NODES * ODIM;
  float* s1     = ws; ws += (size_t)N_NODES * ODIM;

  // --- embedding stage ---
  build_cat_kernel<<<(N_NODES * DCAT + 255) / 256, 256, 0, stream>>>(ids, x, E_all, cat);
  gemm_xwt(cat, qkv_W, qkv_b, qkv, N_NODES, 3 * DCAT, DCAT, stream);
  flash_attn_kernel<<<N_NODES / 64, 128, 0, stream>>>(qkv, att_o);
  gemm_xwt(att_o, proj_W, proj_b, proj_o, N_NODES, DCAT, DCAT, stream);
  gemm_xwt(proj_o, lin_W, lin_b, lin_o, N_NODES, DEMB, DCAT, stream);
  ln_silu_add_kernel<<<N_NODES, DEMB, 0, stream>>>(lin_o, cat, ln1_g, ln1_b, emb0);

  // --- WGAT layer 0 (node-level GEMMs, O(E) edge phase) ---
  gemm_xwt(emb0, Wi0, bi0, hi, N_NODES, HO, DEMB, stream);
  gemm_xwt(emb0, Wj0, bj0, hj, N_NODES, HO, DEMB, stream);
  gemm_xwt(emb0, Wt0, bt0, ht, N_NODES, HO, DEMB, stream);
  alpha_kernel<<<(N_NODES * NHEAD + 255) / 256, 256, 0, stream>>>(hi, attn0, ai);
  alpha_kernel<<<(N_NODES * NHEAD + 255) / 256, 256, 0, stream>>>(hj, attn0, aj);
  edge_logits_kernel<<<NEDGE / 256, 256, 0, stream>>>(eidx, eattr, ai, aj, aedge);
  chunk_softmax_kernel<<<16, 1024, 0, stream>>>(aedge);
  hipMemsetAsync(agg0, 0, (size_t)N_NODES * ODIM * sizeof(float), stream);
  edge_aggregate_kernel<<<NEDGE, ODIM, 0, stream>>>(eidx, aedge, ht, agg0);

  gemm_xwt(agg0, sk0_W, sk0_b, s0, N_NODES, ODIM, ODIM, stream);

  // --- WGAT layer 1 (reuses hi/hj/ht/ai/aj/aedge scratch) ---
  gemm_xwt(agg0, Wi1, bi1, hi, N_NODES, HO, ODIM, stream);
  gemm_xwt(agg0, Wj1, bj1, hj, N_NODES, HO, ODIM, stream);
  gemm_xwt(agg0, Wt1, bt1, ht, N_NODES, HO, ODIM, stream);
  alpha_kernel<<<(N_NODES * NHEAD + 255) / 256, 256, 0, stream>>>(hi, attn1, ai);
  alpha_kernel<<<(N_NODES * NHEAD + 255) / 256, 256, 0, stream>>>(hj, attn1, aj);
  edge_logits_kernel<<<NEDGE / 256, 256, 0, stream>>>(eidx, eattr, ai, aj, aedge);
  chunk_softmax_kernel<<<16, 1024, 0, stream>>>(aedge);
  hipMemsetAsync(agg1, 0, (size_t)N_NODES * ODIM * sizeof(float), stream);
  edge_aggregate_kernel<<<NEDGE, ODIM, 0, stream>>>(eidx, aedge, ht, agg1);

  gemm_xwt(agg1, sk1_W, sk1_b, s1, N_NODES, ODIM, ODIM, stream);

  // --- combine + final LayerNorm ---
  final_ln_kernel<<<N_NODES, ODIM, 0, stream>>>(agg1, s0, s1, nrm_g, nrm_b, out);
}